// Model_51196010168719
// MI455X (gfx1250) — compile-verified
//
#include <hip/hip_runtime.h>
#include <hip/hip_bf16.h>
#include <math.h>

#define IMG_S   128
#define NEAR_P  0.1f
#define FAR_P   100.0f

typedef float v2f __attribute__((ext_vector_type(2)));
typedef float v8f __attribute__((ext_vector_type(8)));

// ---------------------------------------------------------------------------
// Kernel 1: per-face preprocessing.
// For each face f produce:
//   colors[f*4 + 0..2]  lighting color (texture * (amb + dir*relu(n.l)))
//   coef[f*16 + r*4 + k]  four plane equations r = {w0, w1, w2, denom},
//   each (A,B,C,0) so that value(p) = A*px + B*py + C.
//   denom has inv_area/z folded in so zp = 1/(denom + 1e-12).
//   Degenerate faces (|area|<=1e-10) get planes that can never be "inside".
// ---------------------------------------------------------------------------
__device__ __forceinline__ void edge_coef(float ax, float ay, float bx, float by,
                                          float& A, float& B, float& C) {
    float ex = bx - ax, ey = by - ay;
    A = -ey;
    B =  ex;
    C =  ey * ax - ex * ay;
}

__device__ __forceinline__ void project_vert(float px, float py, float pz,
                                             const float* Rm, const float* tv,
                                             const float* Km, float Sf,
                                             float& ox, float& oy, float& oz) {
    // v = R * p + t   (einsum 'bnj,bij->bni')
    float vx = Rm[0]*px + Rm[1]*py + Rm[2]*pz + tv[0];
    float vy = Rm[3]*px + Rm[4]*py + Rm[5]*pz + tv[1];
    float vz = Rm[6]*px + Rm[7]*py + Rm[8]*pz + tv[2];
    float iz = 1.0f / (vz + 1e-9f);
    float x_ = vx * iz, y_ = vy * iz;
    float u  = Km[0]*x_ + Km[1]*y_ + Km[2];
    float v0 = Km[3]*x_ + Km[4]*y_ + Km[5];
    float vv = Sf - v0;
    ox = 2.0f * (u  - 0.5f * Sf) / Sf;
    oy = 2.0f * (vv - 0.5f * Sf) / Sf;
    oz = vz;
}

__device__ __forceinline__ void write_never_inside(float* cp) {
    // w0 = +1, w1 = -1 everywhere -> anyPos && anyNeg -> outside always.
    const float rows[16] = {0,0, 1,0,  0,0,-1,0,  0,0,1,0,  0,0,1,0};
#pragma unroll
    for (int i = 0; i < 16; ++i) cp[i] = rows[i];
}

__global__ void nr_preprocess(const float* __restrict__ vertices,
                              const int*   __restrict__ faces,
                              const float* __restrict__ textures,
                              const float* __restrict__ Rm,
                              const float* __restrict__ tv,
                              const float* __restrict__ Km,
                              const float* __restrict__ lightDir,
                              const float* __restrict__ iDir,
                              const float* __restrict__ iAmb,
                              float* __restrict__ coef,
                              float* __restrict__ colors,
                              int F, int Fpad) {
    int f = blockIdx.x * blockDim.x + threadIdx.x;
    if (f >= Fpad) return;
    float* cp  = coef   + (size_t)f * 16;
    float* col = colors + (size_t)f * 4;
    if (f >= F) {                   // padding face (incl. pipeline prologue pad)
        write_never_inside(cp);
        col[0] = col[1] = col[2] = col[3] = 0.0f;
        return;
    }
    int i0 = faces[f*3+0], i1 = faces[f*3+1], i2 = faces[f*3+2];
    float p0x = vertices[i0*3+0], p0y = vertices[i0*3+1], p0z = vertices[i0*3+2];
    float p1x = vertices[i1*3+0], p1y = vertices[i1*3+1], p1z = vertices[i1*3+2];
    float p2x = vertices[i2*3+0], p2y = vertices[i2*3+1], p2z = vertices[i2*3+2];

    // ---- lighting (world space) ----
    float ax = p0x - p1x, ay = p0y - p1y, az = p0z - p1z;   // v10
    float bx = p2x - p1x, by = p2y - p1y, bz = p2z - p1z;   // v12
    float nx = ay*bz - az*by, ny = az*bx - ax*bz, nz = ax*by - ay*bx;
    float nn = fmaxf(sqrtf(nx*nx + ny*ny + nz*nz), 1e-5f);
    nx /= nn; ny /= nn; nz /= nn;
    float cosT = fmaxf(0.0f, nx*lightDir[0] + ny*lightDir[1] + nz*lightDir[2]);
    float light = iAmb[0] + iDir[0] * cosT;
    col[0] = textures[f*3+0] * light;
    col[1] = textures[f*3+1] * light;
    col[2] = textures[f*3+2] * light;
    col[3] = 0.0f;

    // ---- projection ----
    const float Sf = (float)IMG_S;
    float x0,y0,z0, x1,y1,z1, x2,y2,z2;
    project_vert(p0x,p0y,p0z, Rm,tv,Km,Sf, x0,y0,z0);
    project_vert(p1x,p1y,p1z, Rm,tv,Km,Sf, x1,y1,z1);
    project_vert(p2x,p2y,p2z, Rm,tv,Km,Sf, x2,y2,z2);

    // ---- plane equations ----
    float A0,B0,C0, A1,B1,C1, A2,B2,C2;
    edge_coef(x1,y1, x2,y2, A0,B0,C0);   // w0 = edge(v1,v2)
    edge_coef(x2,y2, x0,y0, A1,B1,C1);   // w1 = edge(v2,v0)
    edge_coef(x0,y0, x1,y1, A2,B2,C2);   // w2 = edge(v0,v1)
    float area = C0 + C1 + C2;           // px/py terms cancel analytically
    if (fabsf(area) <= 1e-10f) {
        write_never_inside(cp);
        return;
    }
    float inv_area = 1.0f / area;
    float iz0 = 1.0f / z0, iz1 = 1.0f / z1, iz2 = 1.0f / z2;
    float dA = inv_area * (A0*iz0 + A1*iz1 + A2*iz2);
    float dB = inv_area * (B0*iz0 + B1*iz1 + B2*iz2);
    float dC = inv_area * (C0*iz0 + C1*iz1 + C2*iz2);

    cp[0]=A0;  cp[1]=B0;  cp[2]=C0;  cp[3]=0.0f;
    cp[4]=A1;  cp[5]=B1;  cp[6]=C1;  cp[7]=0.0f;
    cp[8]=A2;  cp[9]=B2;  cp[10]=C2; cp[11]=0.0f;
    cp[12]=dA; cp[13]=dB; cp[14]=dC; cp[15]=0.0f;
}

// ---------------------------------------------------------------------------
// Kernel 2: WMMA rasterizer.
// One BLOCK (8 waves) handles one 16-pixel tile; each wave scans 1/8 of the
// face groups (groups of 4 faces), so the whole GPU runs 8K waves.
// Per group: D(16x16) = A(16x4 plane coeffs) x B(4x16 pixel basis [px,py,1,0])
// via V_WMMA_F32_16X16X4_F32.
// A layout (ISA 16x4 f32): VGPR0 = K0 (lanes 0-15) / K2 (lanes 16-31),
//                          VGPR1 = K1 / K3, M = lane%16.
// B layout mirrored:  v0 = {px | 1.0}, v1 = {py | 0.0}.
// D layout: lane = pixel column; lanes 0-15 get faces {g*4+0,+1},
//           lanes 16-31 faces {g*4+2,+3}.
// Depth logic is division-free:  zp = 1/d', d' = denom + 1e-12:
//   inside(zp>NEAR && zp<FAR) <=> d' in (1/FAR, 1/NEAR);  argmin zp <=> argmax d'.
// A-operand loads are software-pipelined (coef has 1 extra padded group),
// prefetch is unconditional (speculative prefetch is arch-safe OOB).
// Final merge: half-wave shfl, then an LDS merge of the 8 wave candidates
// in ascending face order (preserves first-min argmin semantics).
// ---------------------------------------------------------------------------
__global__ void nr_raster(const float* __restrict__ coef,
                          const float* __restrict__ colors,
                          float* __restrict__ rgb_out,    // 3*S*S
                          float* __restrict__ gray_out,   // S*S
                          int nGroups) {
    __shared__ float smD[8][16];
    __shared__ int   smI[8][16];

    const int S  = IMG_S;
    const int SS = S * S;
    const int lane   = threadIdx.x & 31;
    const int wave   = threadIdx.x >> 5;
    const int nWaves = blockDim.x >> 5;       // 8
    const int tile   = blockIdx.x;            // one tile per block
    const int pix    = tile * 16 + (lane & 15);

    const int xs = pix % S;
    const int ys = pix / S;
    const float px = (2.0f * (float)xs + 1.0f - (float)S) / (float)S;
    const float py = (2.0f * (float)ys + 1.0f - (float)S) / (float)S;

    const bool hi = (lane >= 16);
    v2f bmat;
    bmat[0] = hi ? 1.0f : px;   // rows K=0 (px) / K=2 (ones)
    bmat[1] = hi ? 0.0f : py;   // rows K=1 (py) / K=3 (zeros)

    const float dLo = 1.0f / FAR_P;    // 0.01
    const float dHi = 1.0f / NEAR_P;   // 10.0

    float bestD  = -INFINITY;          // max d'  ==  min zp
    int   bestIx = -1;

    // face-group chunk for this wave
    const int gPer = (nGroups + nWaves - 1) / nWaves;
    const int gBeg = wave * gPer;
    const int gEnd = (gBeg + gPer < nGroups) ? (gBeg + gPer) : nGroups;

    const float* cbase = coef + (size_t)(lane & 15) * 4 + (hi ? 2 : 0);

    if (gBeg < gEnd) {
        // pipeline prologue: load A for first group
        const float* cp0 = cbase + (size_t)gBeg * 64;
        v2f a;
        a[0] = cp0[0];
        a[1] = cp0[1];

#pragma unroll 2
        for (int g = gBeg; g < gEnd; ++g) {
            // next-iteration A load (coef padded with 1 extra group)
            const float* cpn = cbase + (size_t)(g + 1) * 64;
            v2f an;
            an[0] = cpn[0];
            an[1] = cpn[1];
            __builtin_prefetch(coef + (size_t)(g + 8) * 64, 0, 0);

            v8f c = {};
            c = __builtin_amdgcn_wmma_f32_16x16x4_f32(
                    /*neg_a=*/false, a, /*neg_b=*/false, bmat,
                    /*c_mod=*/(short)0, c, /*reuse_a=*/false, /*reuse_b=*/false);

            const int fbase = g * 4 + (hi ? 2 : 0);
#pragma unroll
            for (int j = 0; j < 2; ++j) {
                float w0 = c[4*j + 0], w1 = c[4*j + 1], w2 = c[4*j + 2];
                float dp = c[4*j + 3] + 1e-12f;
                bool anyNeg = fminf(fminf(w0, w1), w2) < 0.0f;   // v_min3
                bool anyPos = fmaxf(fmaxf(w0, w1), w2) > 0.0f;   // v_max3
                bool inside = !(anyNeg & anyPos) & (dp > dLo) & (dp < dHi);
                if (inside && dp > bestD) { bestD = dp; bestIx = fbase + j; }
            }
            a = an;
        }
    }

    // merge the two half-wave candidates (faces {+0,+1} vs {+2,+3})
    float ob = __shfl_xor(bestD, 16, 32);
    int   oi = __shfl_xor(bestIx, 16, 32);
    if (ob > bestD || (ob == bestD && (unsigned)oi < (unsigned)bestIx)) {
        bestD = ob; bestIx = oi;
    }

    if (lane < 16) {
        smD[wave][lane] = bestD;
        smI[wave][lane] = bestIx;
    }
    __syncthreads();

    // wave 0 merges the per-wave candidates (ascending face order: strict >
    // keeps the earliest face index on depth ties, matching jnp.argmin)
    if (wave == 0 && lane < 16) {
        float bd = smD[0][lane];
        int   bi = smI[0][lane];
        for (int k = 1; k < nWaves; ++k) {
            float d = smD[k][lane];
            int   i = smI[k][lane];
            if (d > bd) { bd = d; bi = i; }
        }
        float r = 0.0f, g = 0.0f, b = 0.0f;
        if (bi >= 0) {
            r = colors[(size_t)bi*4 + 0];
            g = colors[(size_t)bi*4 + 1];
            b = colors[(size_t)bi*4 + 2];
        }
        rgb_out[0*SS + pix] = r;
        rgb_out[1*SS + pix] = g;
        rgb_out[2*SS + pix] = b;
        gray_out[pix] = r + g + b;
    }
}

// ---------------------------------------------------------------------------
// Kernel 3/4: reductions for the loss.
// ---------------------------------------------------------------------------
__global__ void nr_reduce_max(const float* __restrict__ gray,
                              float* __restrict__ redbuf, int n) {
    __shared__ float sm[256];
    float m = -INFINITY;
    for (int i = threadIdx.x; i < n; i += blockDim.x) m = fmaxf(m, gray[i]);
    sm[threadIdx.x] = m;
    __syncthreads();
    for (int s = blockDim.x >> 1; s > 0; s >>= 1) {
        if (threadIdx.x < (unsigned)s)
            sm[threadIdx.x] = fmaxf(sm[threadIdx.x], sm[threadIdx.x + s]);
        __syncthreads();
    }
    if (threadIdx.x == 0) redbuf[0] = sm[0];
}

__global__ void nr_loss(const float* __restrict__ gray,
                        const float* __restrict__ tgt,
                        const float* __restrict__ mask,
                        const float* __restrict__ redbuf,
                        float* __restrict__ out, int n) {
    __shared__ float s0[256];
    __shared__ float s1[256];
    float inv = 1.0f / redbuf[0];
    float a = 0.0f, b = 0.0f;
    for (int i = threadIdx.x; i < n; i += blockDim.x) {
        float d = gray[i] * inv - tgt[i];
        a += d * d;
        b += mask[i];
    }
    s0[threadIdx.x] = a;
    s1[threadIdx.x] = b;
    __syncthreads();
    for (int s = blockDim.x >> 1; s > 0; s >>= 1) {
        if (threadIdx.x < (unsigned)s) {
            s0[threadIdx.x] += s0[threadIdx.x + s];
            s1[threadIdx.x] += s1[threadIdx.x + s];
        }
        __syncthreads();
    }
    if (threadIdx.x == 0) out[0] = s0[0] / s1[0];
}

// ---------------------------------------------------------------------------
extern "C" void kernel_launch(void* const* d_in, const int* in_sizes, int n_in,
                              void* d_out, int out_size, void* d_ws, size_t ws_size,
                              hipStream_t stream) {
    (void)n_in; (void)out_size; (void)ws_size;
    const float* vertices = (const float*)d_in[0];   // (1,N,3)
    const int*   faces    = (const int*  )d_in[1];   // (1,F,3)
    const float* textures = (const float*)d_in[2];   // (1,F,1,1,1,3)
    const float* Rm       = (const float*)d_in[3];   // (1,3,3)
    const float* tv       = (const float*)d_in[4];   // (1,1,3)
    const float* Km       = (const float*)d_in[5];   // (3,3)
    const float* tgt      = (const float*)d_in[6];   // (S,S)
    const float* mask     = (const float*)d_in[7];   // (S,S)
    const float* ldir     = (const float*)d_in[8];   // (3,)
    const float* iDir     = (const float*)d_in[9];   // scalar
    const float* iAmb     = (const float*)d_in[10];  // scalar

    const int S  = IMG_S;
    const int SS = S * S;
    const int F       = in_sizes[1] / 3;
    const int nGroups = (F + 3) / 4;
    const int Fpad    = (nGroups + 1) * 4;           // +1 group: pipeline pad

    float* coef   = (float*)d_ws;                        // Fpad*16 floats
    float* colors = coef + (size_t)Fpad * 16;            // Fpad*4 floats
    float* gray   = colors + (size_t)Fpad * 4;           // SS floats
    float* redbuf = gray + SS;                           // 1 float

    float* loss_out = (float*)d_out;
    float* rgb_out  = loss_out + 1;                      // (3,S,S)

    nr_preprocess<<<(Fpad + 255) / 256, 256, 0, stream>>>(
        vertices, faces, textures, Rm, tv, Km, ldir, iDir, iAmb,
        coef, colors, F, Fpad);

    const int tiles = SS / 16;                           // 1024 blocks
    nr_raster<<<tiles, 256, 0, stream>>>(
        coef, colors, rgb_out, gray, nGroups);

    nr_reduce_max<<<1, 256, 0, stream>>>(gray, redbuf, SS);
    nr_loss<<<1, 256, 0, stream>>>(gray, tgt, mask, redbuf, loss_out, SS);
}